// MPMModel_31035433681623
// MI455X (gfx1250) — compile-verified
//
#include <hip/hip_runtime.h>
#include <hip/hip_bf16.h>
#include <math.h>

// ---------------- MPM constants (from reference) ----------------
#define NGRID 256
#define NG2   (NGRID * NGRID)

#define DT_C      2e-5f
#define INV_DX_C  256.0f
#define DX_F      (1.0f / 256.0f)
#define P_VOL_C   ((DX_F * 0.5f) * (DX_F * 0.5f))
#define P_MASS_C  (P_VOL_C * 1.0f)
#define MU0_C     (1000.0f / (2.0f * (1.0f + 0.2f)))
#define LAM0_C    (1000.0f * 0.2f / ((1.0f + 0.2f) * (1.0f - 2.0f * 0.2f)))
#define GRAV_C    50.0f

typedef unsigned int u32;
typedef __attribute__((ext_vector_type(4))) u32 v4u;
typedef __attribute__((ext_vector_type(8))) int v8i;
typedef __attribute__((ext_vector_type(4))) int v4i;

// ---- TDM: 1-D contiguous copy global -> LDS (D# per CDNA5 ISA) ----
// group0: [count=1 | lds_addr | global_addr lo | global_addr hi + type=2]
// group1: [data_size=4B | tensor_dim0=nelem | tensor_dim1=1 | tile_dim0=nelem |
//          tile_dim1=1 | tensor_dim0_stride=nelem]
__device__ __forceinline__ void tdm_load_1d(const void* gptr, u32 lds_byte_off, u32 nelem)
{
    unsigned long long ga = (unsigned long long)(uintptr_t)gptr;
    v4u g0;
    g0[0] = 1u;                                              // count=1 (valid user D#)
    g0[1] = lds_byte_off;                                    // lds_addr
    g0[2] = (u32)ga;                                         // global_addr[31:0]
    g0[3] = ((u32)(ga >> 32) & 0x01FFFFFFu) | (2u << 30);    // global_addr[56:32] | type=2
    v8i g1;
    g1[0] = (int)(2u << 16);                                 // data_size = 2 (4 bytes)
    g1[1] = (int)((nelem & 0xFFFFu) << 16);                  // tensor_dim0[15:0]
    g1[2] = (int)((nelem >> 16) | (1u << 16));               // tensor_dim0[31:16] | tensor_dim1=1
    g1[3] = (int)((nelem & 0xFFFFu) << 16);                  // tile_dim0 (nelem <= 65535)
    g1[4] = 1;                                               // tile_dim1=1, tile_dim2=0
    g1[5] = (int)nelem;                                      // tensor_dim0_stride[31:0]
    g1[6] = 0;                                               // stride0[47:32]=0, stride1 lo=0
    g1[7] = 0;
    v4i z2 = {0, 0, 0, 0};
    v4i z3 = {0, 0, 0, 0};
#if defined(__clang_major__) && (__clang_major__ >= 23)
    v8i z5 = {0, 0, 0, 0, 0, 0, 0, 0};
    __builtin_amdgcn_tensor_load_to_lds(g0, g1, z2, z3, z5, 0);
#else
    __builtin_amdgcn_tensor_load_to_lds(g0, g1, z2, z3, 0);
#endif
}

// ---------------- grid zero ----------------
__global__ __launch_bounds__(256) void mpm_zero_grid(float* __restrict__ ws, int n) {
    int i = blockIdx.x * blockDim.x + threadIdx.x;
    if (i < n) ws[i] = 0.0f;
}

// ---------------- P2G: particle compute + scatter ----------------
__global__ __launch_bounds__(256) void mpm_p2g(
    const float2* __restrict__ x, const float2* __restrict__ v,
    const float4* __restrict__ C, const float4* __restrict__ F,
    const int*    __restrict__ material, const float* __restrict__ Jp,
    float* __restrict__ gm, float2* __restrict__ gv,
    float4* __restrict__ outF, float* __restrict__ outJp, int* __restrict__ outMat,
    int nPart)
{
    int i = blockIdx.x * blockDim.x + threadIdx.x;
    if (i >= nPart) return;

    float2 xp = x[i];
    float2 vp = v[i];
    float4 Cc = C[i];   // row-major: c00 c01 c10 c11
    float4 Fl = F[i];
    int    mat = material[i];
    float  jp  = Jp[i];

    // F <- F + dt * C @ F
    float f00 = Fl.x + DT_C * (Cc.x * Fl.x + Cc.y * Fl.z);
    float f01 = Fl.y + DT_C * (Cc.x * Fl.y + Cc.y * Fl.w);
    float f10 = Fl.z + DT_C * (Cc.z * Fl.x + Cc.w * Fl.z);
    float f11 = Fl.w + DT_C * (Cc.z * Fl.y + Cc.w * Fl.w);

    float h = __expf(10.0f * (1.0f - jp));
    if (mat == 1) h = 0.3f;
    float mu  = (mat == 0) ? 0.0f : MU0_C * h;
    float lam = LAM0_C * h;

    // ---- closed-form 2x2 SVD:  A = U diag(s1,s2) V^T ----
    float Es = 0.5f * (f00 + f11);
    float Fs = 0.5f * (f00 - f11);
    float Gs = 0.5f * (f10 + f01);
    float Hs = 0.5f * (f10 - f01);
    float Qv = sqrtf(Es * Es + Hs * Hs);
    float Rv = sqrtf(Fs * Fs + Gs * Gs);
    float s1 = Qv + Rv;
    float s2 = Qv - Rv;
    float a1 = atan2f(Gs, Fs);
    float a2 = atan2f(Hs, Es);
    float phi = 0.5f * (a2 + a1);   // U angle
    float th  = 0.5f * (a1 - a2);   // V angle
    float su, cu, sv, cv;
    __sincosf(phi, &su, &cu);
    __sincosf(th,  &sv, &cv);
    // fold sign of s2 into V^T row 1 so singular values are non-negative
    float sgn  = (s2 < 0.0f) ? -1.0f : 1.0f;
    float sig1 = s1;
    float sig2 = fabsf(s2);
    float vh00 = cv,         vh01 = sv;
    float vh10 = -sv * sgn,  vh11 = cv * sgn;

    // snow plasticity: clamp singular values, track Jp
    if (mat == 2) {
        float c1 = fminf(fmaxf(sig1, 1.0f - 2.5e-2f), 1.0f + 4.5e-3f);
        float c2 = fminf(fmaxf(sig2, 1.0f - 2.5e-2f), 1.0f + 4.5e-3f);
        jp *= (sig1 / c1) * (sig2 / c2);
        sig1 = c1;
        sig2 = c2;
    }
    float J = sig1 * sig2;

    // F reconstruction by material
    if (mat == 0) {               // fluid: sqrt(J) * I
        float s = sqrtf(J);
        f00 = s; f01 = 0.0f; f10 = 0.0f; f11 = s;
    } else if (mat == 2) {        // snow: U diag(sig) V^T
        float us00 = cu * sig1, us01 = -su * sig2;
        float us10 = su * sig1, us11 =  cu * sig2;
        f00 = us00 * vh00 + us01 * vh10;
        f01 = us00 * vh01 + us01 * vh11;
        f10 = us10 * vh00 + us11 * vh10;
        f11 = us10 * vh01 + us11 * vh11;
    }

    // R = U @ V^T
    float r00 = cu * vh00 - su * vh10;
    float r01 = cu * vh01 - su * vh11;
    float r10 = su * vh00 + cu * vh10;
    float r11 = su * vh01 + cu * vh11;

    // stress = 2*mu*(F-R)F^T + lam*J*(J-1)*I, scaled; affine = stress + m*C
    float d00 = f00 - r00, d01 = f01 - r01, d10 = f10 - r10, d11 = f11 - r11;
    float tm   = 2.0f * mu;
    float diag = lam * J * (J - 1.0f);
    float st00 = tm * (d00 * f00 + d01 * f01) + diag;
    float st01 = tm * (d00 * f10 + d01 * f11);
    float st10 = tm * (d10 * f00 + d11 * f01);
    float st11 = tm * (d10 * f10 + d11 * f11) + diag;
    float k   = -DT_C * P_VOL_C * 4.0f * INV_DX_C * INV_DX_C;
    float a00 = k * st00 + P_MASS_C * Cc.x;
    float a01 = k * st01 + P_MASS_C * Cc.y;
    float a10 = k * st10 + P_MASS_C * Cc.z;
    float a11 = k * st11 + P_MASS_C * Cc.w;

    // quadratic B-spline weights
    float px = xp.x * INV_DX_C, py = xp.y * INV_DX_C;
    int bx = (int)floorf(px - 0.5f);
    int by = (int)floorf(py - 0.5f);
    float fxx = px - (float)bx, fxy = py - (float)by;
    float wxa[3], wya[3];
    wxa[0] = 0.5f * (1.5f - fxx) * (1.5f - fxx);
    wxa[1] = 0.75f - (fxx - 1.0f) * (fxx - 1.0f);
    wxa[2] = 0.5f * (fxx - 0.5f) * (fxx - 0.5f);
    wya[0] = 0.5f * (1.5f - fxy) * (1.5f - fxy);
    wya[1] = 0.75f - (fxy - 1.0f) * (fxy - 1.0f);
    wya[2] = 0.5f * (fxy - 0.5f) * (fxy - 0.5f);

    float mvx = P_MASS_C * vp.x;
    float mvy = P_MASS_C * vp.y;

    // scatter momentum (fused: w*(m*v + affine@dpos)) + mass; 3 f32 atomics/tap
#pragma unroll
    for (int ii = 0; ii < 3; ++ii) {
        int gx = bx + ii;
        float dpx = ((float)ii - fxx) * DX_F;
#pragma unroll
        for (int jj = 0; jj < 3; ++jj) {
            int gy = by + jj;
            if ((unsigned)gx < (unsigned)NGRID && (unsigned)gy < (unsigned)NGRID) {
                float dpy = ((float)jj - fxy) * DX_F;
                float wt = wxa[ii] * wya[jj];
                int node = gx * NGRID + gy;
                atomicAdd(&gv[node].x, wt * (mvx + a00 * dpx + a01 * dpy));
                atomicAdd(&gv[node].y, wt * (mvy + a10 * dpx + a11 * dpy));
                atomicAdd(&gm[node],   wt * P_MASS_C);
            }
        }
    }

    // grid-independent outputs
    float4 fo; fo.x = f00; fo.y = f01; fo.z = f10; fo.w = f11;
    outF[i]   = fo;
    outJp[i]  = jp;
    outMat[i] = mat;
}

// ---------------- grid ops: TDM-stage to LDS, momentum->velocity, gravity, walls ----------------
__global__ __launch_bounds__(256) void mpm_grid(
    const float* __restrict__ gm, float2* __restrict__ gv)
{
    __shared__ float s_gm[256];
    __shared__ float s_gv[512];

    int t    = threadIdx.x;
    int base = blockIdx.x * 256;

    // wave 0 issues two TDM 1-D tile copies (TENSORcnt-tracked), others wait at barrier
    if (t < 32) {
        tdm_load_1d(gm + base, (u32)(uintptr_t)(void*)s_gm, 256u);
        tdm_load_1d((const float*)gv + 2 * base, (u32)(uintptr_t)(void*)s_gv, 512u);
        __builtin_amdgcn_s_wait_tensorcnt(0);
    }
    __syncthreads();

    int n = base + t;
    float m  = s_gm[t];
    float vx = s_gv[2 * t + 0];
    float vy = s_gv[2 * t + 1];
    if (m > 0.0f) {
        float inv = 1.0f / fmaxf(m, 1e-12f);
        vx *= inv;
        vy *= inv;
    }
    vy -= DT_C * GRAV_C;
    int i = n >> 8;      // grid row (x index)
    int j = n & 255;     // grid col (y index)
    if (i < 3)           vx = fmaxf(vx, 0.0f);
    if (i >= NGRID - 2)  vx = fminf(vx, 0.0f);
    if (j < 3)           vy = fmaxf(vy, 0.0f);
    if (j >= NGRID - 2)  vy = fminf(vy, 0.0f);
    float2 o; o.x = vx; o.y = vy;
    gv[n] = o;
}

// ---------------- G2P: gather, new v/C, advect ----------------
__global__ __launch_bounds__(256) void mpm_g2p(
    const float2* __restrict__ x,
    const float2* __restrict__ gv,
    float2* __restrict__ outX, float2* __restrict__ outV,
    float4* __restrict__ outC, int nPart)
{
    int i = blockIdx.x * blockDim.x + threadIdx.x;
    if (i >= nPart) return;

    float2 xp = x[i];
    float px = xp.x * INV_DX_C, py = xp.y * INV_DX_C;
    int bx = (int)floorf(px - 0.5f);
    int by = (int)floorf(py - 0.5f);
    float fxx = px - (float)bx, fxy = py - (float)by;

    // pre-touch the 3 grid rows while weight math is in flight (global_prefetch_b8)
#pragma unroll
    for (int ii = 0; ii < 3; ++ii) {
        int row = min(max(bx + ii, 0), NGRID - 1) * NGRID + min(max(by, 0), NGRID - 1);
        __builtin_prefetch(&gv[row], 0, 3);
    }

    float wxa[3], wya[3];
    wxa[0] = 0.5f * (1.5f - fxx) * (1.5f - fxx);
    wxa[1] = 0.75f - (fxx - 1.0f) * (fxx - 1.0f);
    wxa[2] = 0.5f * (fxx - 0.5f) * (fxx - 0.5f);
    wya[0] = 0.5f * (1.5f - fxy) * (1.5f - fxy);
    wya[1] = 0.75f - (fxy - 1.0f) * (fxy - 1.0f);
    wya[2] = 0.5f * (fxy - 0.5f) * (fxy - 0.5f);

    float nvx = 0.0f, nvy = 0.0f;
    float b00 = 0.0f, b01 = 0.0f, b10 = 0.0f, b11 = 0.0f;

#pragma unroll
    for (int ii = 0; ii < 3; ++ii) {
        float dpx = ((float)ii - fxx) * DX_F;
#pragma unroll
        for (int jj = 0; jj < 3; ++jj) {
            float dpy = ((float)jj - fxy) * DX_F;
            float wt = wxa[ii] * wya[jj];
            // clamp flat index (matches jnp gather clamping on OOB)
            int node = (bx + ii) * NGRID + (by + jj);
            node = min(max(node, 0), NG2 - 1);
            float2 g = gv[node];                 // one global_load_b64 per tap
            nvx += wt * g.x;
            nvy += wt * g.y;
            b00 += wt * g.x * dpx;  b01 += wt * g.x * dpy;
            b10 += wt * g.y * dpx;  b11 += wt * g.y * dpy;
        }
    }

    float sc = 4.0f * INV_DX_C * INV_DX_C;
    float4 co; co.x = sc * b00; co.y = sc * b01; co.z = sc * b10; co.w = sc * b11;
    float2 vo; vo.x = nvx; vo.y = nvy;
    float2 xo; xo.x = xp.x + DT_C * nvx; xo.y = xp.y + DT_C * nvy;

    outC[i] = co;
    outV[i] = vo;
    outX[i] = xo;
}

// ---------------- launcher ----------------
extern "C" void kernel_launch(void* const* d_in, const int* in_sizes, int n_in,
                              void* d_out, int out_size, void* d_ws, size_t ws_size,
                              hipStream_t stream)
{
    const float2* x   = (const float2*)d_in[0];
    const float2* v   = (const float2*)d_in[1];
    const float4* C   = (const float4*)d_in[2];
    const float4* F   = (const float4*)d_in[3];
    const int*    mat = (const int*)   d_in[4];
    const float*  Jp  = (const float*) d_in[5];

    const int N = in_sizes[0] / 2;   // particles

    float*  ws = (float*)d_ws;
    float*  gm = ws;                        // [NG2]
    float2* gv = (float2*)(ws + NG2);       // [NG2] interleaved (vx, vy)

    float* out   = (float*)d_out;
    float2* oX   = (float2*)(out);            // [N,2]
    float2* oV   = (float2*)(out + 2 * N);    // [N,2]
    float4* oC   = (float4*)(out + 4 * N);    // [N,2,2]
    float4* oF   = (float4*)(out + 8 * N);    // [N,2,2]
    int*    oMat = (int*)   (out + 12 * N);   // [N] (bit-copied int32)
    float*  oJp  =          (out + 13 * N);   // [N]

    const int TB = 256;

    mpm_zero_grid<<<(3 * NG2 + TB - 1) / TB, TB, 0, stream>>>(ws, 3 * NG2);

    mpm_p2g<<<(N + TB - 1) / TB, TB, 0, stream>>>(
        x, v, C, F, mat, Jp, gm, gv, oF, oJp, oMat, N);

    mpm_grid<<<NG2 / TB, TB, 0, stream>>>(gm, gv);

    mpm_g2p<<<(N + TB - 1) / TB, TB, 0, stream>>>(
        x, gv, oX, oV, oC, N);
}